// SSIMLoss_66709432042083
// MI455X (gfx1250) — compile-verified
//
#include <hip/hip_runtime.h>

typedef __attribute__((ext_vector_type(2))) float v2f;
typedef __attribute__((ext_vector_type(8))) float v8f;

#define SSIM_C1 1.0e-4f
#define SSIM_C2 9.0e-4f

#define IMG_H 512
#define IMG_W 512
#define PLANE (IMG_H * IMG_W)

#define TILE_ROWS 64      // per workgroup
#define TILE_COLS 32      // per workgroup
#define IN_ROWS 80        // halo rows loaded (covers y0-5 .. y0+74)
#define IN_COLS 44        // halo cols loaded (covers x0-5 .. x0+38)
#define IN_STRIDE 45      // padded to avoid 16-row bank conflicts
#define MID_STRIDE 17     // padded mid scratch stride

#define NTHREADS 128      // 4 waves; each wave computes 2 16x16 output tiles

// D = A(16x4) * B(4x16) + C, fp32 WMMA, chained over K
static __device__ __forceinline__ v8f wmma4(v2f a, v2f b, v8f c) {
    return __builtin_amdgcn_wmma_f32_16x16x4_f32(
        /*neg_a=*/false, a, /*neg_b=*/false, b,
        /*c_mod=*/(short)0, c, /*reuse_a=*/false, /*reuse_b=*/false);
}

__global__ __launch_bounds__(NTHREADS) void ssim_tile_kernel(
    const float* __restrict__ img1, const float* __restrict__ img2,
    float* __restrict__ partial)
{
    __shared__ float s1[IN_ROWS * IN_STRIDE];
    __shared__ float s2[IN_ROWS * IN_STRIDE];
    __shared__ float smid[4][5][16 * MID_STRIDE];  // per-wave, per-quantity half-mid
    __shared__ float sred[NTHREADS];

    const int tid  = threadIdx.x;
    const int lane = tid & 31;
    const int wid  = tid >> 5;
    const int half = lane >> 4;     // 0: lanes 0-15, 1: lanes 16-31
    const int l16  = lane & 15;

    const int bid   = blockIdx.x;
    const int plane = bid >> 7;               // 128 tiles per 512x512 plane
    const int t     = bid & 127;
    const int wgY   = (t >> 4) * TILE_ROWS;   // 8 row tiles
    const int wgX   = (t & 15) * TILE_COLS;   // 16 col tiles
    const float* __restrict__ p1 = img1 + (size_t)plane * PLANE;
    const float* __restrict__ p2 = img2 + (size_t)plane * PLANE;

    // ---- cooperative zero-padded halo load ----
    for (int i = tid; i < IN_ROWS * IN_COLS; i += NTHREADS) {
        int r = i / IN_COLS;
        int c = i - r * IN_COLS;
        int y = wgY - 5 + r;
        int x = wgX - 5 + c;
        float a = 0.f, b = 0.f;
        if ((unsigned)y < IMG_H && (unsigned)x < IMG_W) {
            a = p1[y * IMG_W + x];
            b = p2[y * IMG_W + x];
        }
        s1[r * IN_STRIDE + c] = a;
        s2[r * IN_STRIDE + c] = b;
    }
    __syncthreads();

    // ---- per-lane banded Gaussian weight slices (sigma=1.5, 11 taps) ----
    // Serves as B for horizontal pass and as A for vertical pass:
    // value = g[K - (lane&15)], K = 4*step + 2*half + j, zero outside [0,10].
    float S = 0.f;
#pragma unroll
    for (int i = 0; i < 11; ++i) {
        float d = (float)(i - 5);
        S += __expf(-d * d * (1.0f / 4.5f));
    }
    float invS = 1.0f / S;
    v2f wsl[7];
#pragma unroll
    for (int s = 0; s < 7; ++s) {
#pragma unroll
        for (int j = 0; j < 2; ++j) {
            int K = 4 * s + 2 * half + j;
            int d = K - l16;
            float w = 0.f;
            if (d >= 0 && d <= 10) {
                float e = (float)(d - 5);
                w = __expf(-e * e * (1.0f / 4.5f)) * invS;
            }
            wsl[s][j] = w;
        }
    }

    float local = 0.f;

    // ---- each wave computes two 16x16 output tiles ----
#pragma unroll
    for (int tt = 0; tt < 2; ++tt) {
        const int tile = wid * 2 + tt;        // 0..7
        const int y0 = (tile >> 1) * 16;      // 0,16,32,48
        const int x0 = (tile & 1) * 16;       // 0,16

        v8f Dacc[5];
#pragma unroll
        for (int q = 0; q < 5; ++q) Dacc[q] = (v8f){};

        // Two 16-row horizontal halves; vertical accumulation interleaved:
        // vertical steps 0..3 consume mid rows 0..15 (r=0),
        // vertical steps 4..6 consume mid rows 16..27 (r=1).
#pragma unroll
        for (int r = 0; r < 2; ++r) {
            v8f C[5];
#pragma unroll
            for (int q = 0; q < 5; ++q) C[q] = (v8f){};

            const int rrow = y0 + 16 * r + l16;   // shifted LDS row index
#pragma unroll
            for (int s = 0; s < 7; ++s) {
                int cc = x0 + 4 * s + 2 * half;
                float x1a = s1[rrow * IN_STRIDE + cc];
                float x1b = s1[rrow * IN_STRIDE + cc + 1];
                float x2a = s2[rrow * IN_STRIDE + cc];
                float x2b = s2[rrow * IN_STRIDE + cc + 1];
                v2f a;
                a[0] = x1a;       a[1] = x1b;       C[0] = wmma4(a, wsl[s], C[0]);
                a[0] = x2a;       a[1] = x2b;       C[1] = wmma4(a, wsl[s], C[1]);
                a[0] = x1a * x1a; a[1] = x1b * x1b; C[2] = wmma4(a, wsl[s], C[2]);
                a[0] = x2a * x2a; a[1] = x2b * x2b; C[3] = wmma4(a, wsl[s], C[3]);
                a[0] = x1a * x2a; a[1] = x1b * x2b; C[4] = wmma4(a, wsl[s], C[4]);
            }

            // D-layout -> half-mid LDS (local row = v + 8*half)
#pragma unroll
            for (int q = 0; q < 5; ++q) {
#pragma unroll
                for (int v = 0; v < 8; ++v)
                    smid[wid][q][(v + 8 * half) * MID_STRIDE + l16] = C[q][v];
            }

            // Vertical partial pass for this half: D += Wv_slice * mid_half
            const int sbeg = (r == 0) ? 0 : 4;
            const int send = (r == 0) ? 4 : 7;
#pragma unroll
            for (int s = sbeg; s < send; ++s) {
                int kb = 4 * s + 2 * half - 16 * r;   // local mid row base
#pragma unroll
                for (int q = 0; q < 5; ++q) {
                    v2f b;
                    b[0] = smid[wid][q][(kb + 0) * MID_STRIDE + l16];
                    b[1] = smid[wid][q][(kb + 1) * MID_STRIDE + l16];
                    Dacc[q] = wmma4(wsl[s], b, Dacc[q]);
                }
            }
        }

        // ---- per-pixel SSIM on the 5 accumulators (8 pixels per lane) ----
#pragma unroll
        for (int v = 0; v < 8; ++v) {
            float m1  = Dacc[0][v];
            float m2  = Dacc[1][v];
            float e11 = Dacc[2][v];
            float e22 = Dacc[3][v];
            float e12 = Dacc[4][v];
            float m1s = m1 * m1, m2s = m2 * m2, m12 = m1 * m2;
            float sg1 = e11 - m1s, sg2 = e22 - m2s, sg12 = e12 - m12;
            float num = (2.f * m12 + SSIM_C1) * (2.f * sg12 + SSIM_C2);
            float den = (m1s + m2s + SSIM_C1) * (sg1 + sg2 + SSIM_C2);
            local += num / den;
        }
    }

    // ---- deterministic block reduction -> per-block partial ----
    sred[tid] = local;
    __syncthreads();
#pragma unroll
    for (int s = NTHREADS / 2; s > 0; s >>= 1) {
        if (tid < s) sred[tid] += sred[tid + s];
        __syncthreads();
    }
    if (tid == 0) partial[bid] = sred[0];
}

__global__ __launch_bounds__(256) void ssim_finalize_kernel(
    const float* __restrict__ partial, float* __restrict__ out, int n)
{
    __shared__ float sr[256];
    float acc = 0.f;
    for (int i = threadIdx.x; i < n; i += 256) acc += partial[i];
    sr[threadIdx.x] = acc;
    __syncthreads();
#pragma unroll
    for (int s = 128; s > 0; s >>= 1) {
        if (threadIdx.x < s) sr[threadIdx.x] += sr[threadIdx.x + s];
        __syncthreads();
    }
    if (threadIdx.x == 0)
        out[0] = 1.0f - sr[0] * (1.0f / 25165824.0f);   // N = 32*3*512*512
}

extern "C" void kernel_launch(void* const* d_in, const int* in_sizes, int n_in,
                              void* d_out, int out_size, void* d_ws, size_t ws_size,
                              hipStream_t stream) {
    (void)in_sizes; (void)n_in; (void)out_size; (void)ws_size;
    const float* img1 = (const float*)d_in[0];
    const float* img2 = (const float*)d_in[1];
    float* out = (float*)d_out;
    float* partial = (float*)d_ws;

    const int nblocks = 32 * 3 * (IMG_H / TILE_ROWS) * (IMG_W / TILE_COLS); // 12288
    ssim_tile_kernel<<<nblocks, NTHREADS, 0, stream>>>(img1, img2, partial);
    ssim_finalize_kernel<<<1, 256, 0, stream>>>(partial, out, nblocks);
}